// AlphaModel_67516885893391
// MI455X (gfx1250) — compile-verified
//
#include <hip/hip_runtime.h>
#include <math.h>

#define TPB 256

typedef int v4i __attribute__((vector_size(16)));
typedef __attribute__((address_space(1))) v4i as1_v4i;   // global 16B chunk
typedef __attribute__((address_space(3))) v4i as3_v4i;   // LDS 16B chunk

struct F3 { float x, y, z; };   // 12-byte packed stream element -> global_load_b96

__global__ __launch_bounds__(TPB) void alpha_model_kernel(
    const float* __restrict__ prnt,    // [n*3]
    const float* __restrict__ child,   // [n*3]
    const int*   __restrict__ rels,    // [n]
    const float* __restrict__ M,       // [64*9]
    const float* __restrict__ beta,    // [64*3]
    const float* __restrict__ zeps_p,  // [1]
    const float* __restrict__ sf_p,    // [1]
    float*       __restrict__ out,     // [n*3]
    int n)
{
    __shared__ float sM[576];   // 64 relations x 9
    __shared__ float sB[192];   // 64 relations x 3

    const int t = threadIdx.x;

    // ---- Stage the 3KB parameter tables into LDS (CDNA5 async copy path) ----
#if defined(__HIP_DEVICE_COMPILE__) && __has_builtin(__builtin_amdgcn_global_load_async_to_lds_b128)
    if (t < 192) {
        if (t < 144) {
            // 144 * 16B = 2304B = all of M
            __builtin_amdgcn_global_load_async_to_lds_b128(
                (as1_v4i*)(M + 4 * t), (as3_v4i*)(sM + 4 * t), 0, 0);
        } else {
            // 48 * 16B = 768B = all of beta
            const int u = t - 144;
            __builtin_amdgcn_global_load_async_to_lds_b128(
                (as1_v4i*)(beta + 4 * u), (as3_v4i*)(sB + 4 * u), 0, 0);
        }
    }
#if __has_builtin(__builtin_amdgcn_s_wait_asynccnt)
    __builtin_amdgcn_s_wait_asynccnt(0);
#else
    asm volatile("s_wait_asynccnt 0" ::: "memory");
#endif
#else
    // Fallback path (also what the host-side parse sees)
    for (int i = t; i < 576; i += TPB) sM[i] = M[i];
    for (int i = t; i < 192; i += TPB) sB[i] = beta[i];
#endif
    __syncthreads();

    const float zeps = zeps_p[0];
    const float sf   = sf_p[0];

    const F3* __restrict__ p3 = (const F3*)prnt;
    const F3* __restrict__ c3 = (const F3*)child;
    F3*       __restrict__ o3 = (F3*)out;

    const int stride = (int)(gridDim.x) * TPB;
    for (int e = (int)(blockIdx.x) * TPB + t; e < n; e += stride) {
        // Prefetch next grid-stride iteration's stream data (clamped in-range).
        {
            int ep = e + stride;
            ep = (ep < n) ? ep : e;
            __builtin_prefetch(prnt  + 3 * (size_t)ep, 0, 0);
            __builtin_prefetch(child + 3 * (size_t)ep, 0, 0);
        }

        const F3 p = p3[e];            // global_load_b96
        const F3 c = c3[e];            // global_load_b96
        const int r = rels[e];         // global_load_b32

        const float* __restrict__ mr = sM + r * 9;  // ds_load gathers
        const float* __restrict__ br = sB + r * 3;

        // scores = M[r] @ child_probs
        const float s0 = fmaf(mr[0], c.x, fmaf(mr[1], c.y, mr[2] * c.z));
        const float s1 = fmaf(mr[3], c.x, fmaf(mr[4], c.y, mr[5] * c.z));
        const float s2 = fmaf(mr[6], c.x, fmaf(mr[7], c.y, mr[8] * c.z));

        // softmax (max-subtracted, like jax.nn.softmax)
        const float mx = fmaxf(s0, fmaxf(s1, s2));
        const float e0 = expf(s0 - mx);
        const float e1 = expf(s1 - mx);
        const float e2 = expf(s2 - mx);
        const float inv = 1.0f / (e0 + e1 + e2);
        const float chx = e0 * inv, chy = e1 * inv, chz = e2 * inv;

        // entropy of renormalized clipped blend
        const float zx = fmaxf(zeps, p.x + chx);
        const float zy = fmaxf(zeps, p.y + chy);
        const float zz = fmaxf(zeps, p.z + chz);
        const float iZ = 1.0f / (zx + zy + zz);
        const float nx = zx * iZ, ny = zy * iZ, nz = zz * iZ;
        const float ent = -(nx * logf(nx) + ny * logf(ny) + nz * logf(nz));

        // cosine similarity between prnt and softmax(child)
        const float dotpc = p.x * chx + p.y * chy + p.z * chz;
        float nrm = sqrtf((p.x * p.x + p.y * p.y + p.z * p.z) *
                          (chx * chx + chy * chy + chz * chz));
        nrm = (nrm == 0.0f) ? 1.0f : nrm;
        const float cosv = 1.1f + dotpc / nrm;
        const float scale = sf * cosv / ent;

        // alpha = (1-b)*prnt + b*child, scaled
        const float b0 = br[0], b1 = br[1], b2 = br[2];
        F3 o;
        o.x = fmaf(b0, chx - p.x, p.x) * scale;  // (1-b)*p + b*ch == p + b*(ch-p)
        o.y = fmaf(b1, chy - p.y, p.y) * scale;
        o.z = fmaf(b2, chz - p.z, p.z) * scale;
        o3[e] = o;                     // global_store_b96
    }
}

extern "C" void kernel_launch(void* const* d_in, const int* in_sizes, int n_in,
                              void* d_out, int out_size, void* d_ws, size_t ws_size,
                              hipStream_t stream) {
    // setup_inputs order:
    // 0: var_sfx (int scalar, unused)
    // 1: prnt_probs f32 [E,3]   2: child_probs f32 [E,3]   3: rels i32 [E]
    // 4: M f32 [64,3,3]         5: beta f32 [64,3]
    // 6: z_epsilon f32 [1]      7: scale_factor f32 [1]
    const float* prnt  = (const float*)d_in[1];
    const float* child = (const float*)d_in[2];
    const int*   rels  = (const int*)d_in[3];
    const float* M     = (const float*)d_in[4];
    const float* beta  = (const float*)d_in[5];
    const float* zeps  = (const float*)d_in[6];
    const float* sf    = (const float*)d_in[7];
    float* out = (float*)d_out;

    const int n = in_sizes[3];             // number of edges
    int nblk = (n + TPB - 1) / TPB;
    if (nblk > 4096) nblk = 4096;          // grid-stride; ~10 edges/thread
    if (nblk < 1)    nblk = 1;

    alpha_model_kernel<<<nblk, TPB, 0, stream>>>(prnt, child, rels, M, beta,
                                                 zeps, sf, out, n);
}